// GCNII_5179730559510
// MI455X (gfx1250) — compile-verified
//
#include <hip/hip_runtime.h>
#include <hip/hip_bf16.h>
#include <math.h>

typedef __attribute__((ext_vector_type(2))) float v2f;
typedef __attribute__((ext_vector_type(8))) float v8f;

// -------------------------------------------------------------------------
// Generic fused WMMA f32 GEMM:
//   C = maybe_relu( alpha * (A @ B) + bias + beta * S ), optional dup-store C2
// A: [M,K] row-major, B: [K,Ncols] row-major, S: [M,256] row-major.
// Block = 256 threads = 8 waves; workgroup tile 128x64; wave tile 16x64.
// B tile staged K-pair-interleaved so each B fragment is one ds_load_b64.
// -------------------------------------------------------------------------
__global__ __launch_bounds__(256, 2)
void gcnii_wmma_gemm_f32(const float* __restrict__ A, const float* __restrict__ B,
                         const float* __restrict__ bias, const float* __restrict__ S,
                         float* __restrict__ C, float* __restrict__ C2,
                         int M, int K, int Ncols, float alpha, float beta, int relu)
{
    __shared__ __align__(16) float As[128][36];    // 32 cols + pad (stride 144B)
    // B (k,col) stored at BsP[k>>1][col*2 + (k&1)]; stride 160 f32 (== 32 mod 64
    // banks) -> half-waves land on disjoint bank halves for b64 fragment reads.
    __shared__ __align__(16) float BsP[16][160];

    const int tid   = threadIdx.x;
    const int wave  = tid >> 5;
    const int lane  = tid & 31;
    const int lmod  = lane & 15;
    const int lhalf = lane >> 4;
    const int tileM = blockIdx.y * 128;
    const int tileN = blockIdx.x * 64;

    v8f acc0 = {}, acc1 = {}, acc2 = {}, acc3 = {};

    for (int k0 = 0; k0 < K; k0 += 32) {
        // ---- stage A tile [128 x 32] (coalesced float4 loads, M-guarded) ----
        {
            const int quad  = tid & 7;      // which 16B chunk of the 128B row-slab
            const int rbase = tid >> 3;     // 0..31
#pragma unroll
            for (int i = 0; i < 4; ++i) {
                const int row = rbase + i * 32;
                const int gr  = tileM + row;
                float4 v = make_float4(0.f, 0.f, 0.f, 0.f);
                if (gr < M) {
                    const float* ap = A + (size_t)gr * K + k0 + quad * 4;
                    v = *(const float4*)ap;
                    if (k0 + 32 < K) __builtin_prefetch(ap + 32, 0, 1);
                }
                *(float4*)&As[row][quad * 4] = v;
            }
        }
        // ---- stage B tile [32 x 64], K-pair interleaved (Ncols-guarded) ----
        {
            const int col4  = (tid & 15) * 4;
            const int rbase = tid >> 4;     // 0..15
#pragma unroll
            for (int i = 0; i < 2; ++i) {
                const int krow = rbase + i * 16;
                const float* bp = B + (size_t)(k0 + krow) * Ncols;
                float* dst = &BsP[krow >> 1][(krow & 1)];
#pragma unroll
                for (int j = 0; j < 4; ++j) {
                    const int gc = tileN + col4 + j;
                    dst[(col4 + j) * 2] = (gc < Ncols) ? bp[gc] : 0.f;
                }
            }
        }
        __syncthreads();

        // ---- 8 WMMA steps over the 32-deep slab, 4 N-tiles each ----
#pragma unroll
        for (int kk = 0; kk < 32; kk += 4) {
            const int kr = kk + 2 * lhalf;           // even
            v2f a;
            a.x = As[wave * 16 + lmod][kr + 0];
            a.y = As[wave * 16 + lmod][kr + 1];
            const float* brow = &BsP[kr >> 1][0];
            v2f b0 = *(const v2f*)(brow + ( 0 + lmod) * 2);
            v2f b1 = *(const v2f*)(brow + (16 + lmod) * 2);
            v2f b2 = *(const v2f*)(brow + (32 + lmod) * 2);
            v2f b3 = *(const v2f*)(brow + (48 + lmod) * 2);
            acc0 = __builtin_amdgcn_wmma_f32_16x16x4_f32(false, a, false, b0, (short)0, acc0, false, false);
            acc1 = __builtin_amdgcn_wmma_f32_16x16x4_f32(false, a, false, b1, (short)0, acc1, false, false);
            acc2 = __builtin_amdgcn_wmma_f32_16x16x4_f32(false, a, false, b2, (short)0, acc2, false, false);
            acc3 = __builtin_amdgcn_wmma_f32_16x16x4_f32(false, a, false, b3, (short)0, acc3, false, false);
        }
        __syncthreads();
    }

    // ---- fused epilogue: alpha*AB + bias + beta*S, relu, (dup) store ----
    const int rowBase = tileM + wave * 16 + 8 * lhalf;   // C/D layout: M = r + 8*(lane/16)
    v8f accs[4] = {acc0, acc1, acc2, acc3};
#pragma unroll
    for (int t = 0; t < 4; ++t) {
        const int col = tileN + t * 16 + lmod;
        if (col >= Ncols) continue;
        const float bv = bias ? bias[col] : 0.f;
#pragma unroll
        for (int r = 0; r < 8; ++r) {
            const int row = rowBase + r;
            if (row >= M) break;
            float v = alpha * accs[t][r] + bv;
            if (S) v += beta * S[(size_t)row * 256 + col];
            if (relu) v = v > 0.f ? v : 0.f;
            C[(size_t)row * Ncols + col] = v;
            if (C2) C2[(size_t)row * Ncols + col] = v;
        }
    }
}

// -------------------------------------------------------------------------
// Edge-parallel SpMM scatter: one wave32 per edge; lane handles 8 floats.
// h is 102 MB -> L2-resident on MI455X (192 MB L2): gathers hit L2.
// -------------------------------------------------------------------------
__global__ __launch_bounds__(256)
void gcnii_spmm_edges(const int* __restrict__ erow, const int* __restrict__ ecol,
                      const float* __restrict__ eval, const float* __restrict__ h,
                      float* __restrict__ hi, int E)
{
    const int w = (int)((blockIdx.x * 256u + threadIdx.x) >> 5);
    if (w >= E) return;
    const int lane = threadIdx.x & 31;
    const int r = erow[w];
    const int c = ecol[w];
    const float v = eval[w];

    const float4* src = (const float4*)(h + (size_t)c * 256) + lane * 2;
    const float4 x0 = src[0];
    const float4 x1 = src[1];

    float* dst = hi + (size_t)r * 256 + lane * 8;
    unsafeAtomicAdd(dst + 0, v * x0.x);
    unsafeAtomicAdd(dst + 1, v * x0.y);
    unsafeAtomicAdd(dst + 2, v * x0.z);
    unsafeAtomicAdd(dst + 3, v * x0.w);
    unsafeAtomicAdd(dst + 4, v * x1.x);
    unsafeAtomicAdd(dst + 5, v * x1.y);
    unsafeAtomicAdd(dst + 6, v * x1.z);
    unsafeAtomicAdd(dst + 7, v * x1.w);
}

__global__ __launch_bounds__(256)
void gcnii_zero_f4(float4* __restrict__ p, int n4)
{
    const int i = blockIdx.x * 256 + threadIdx.x;
    if (i < n4) p[i] = make_float4(0.f, 0.f, 0.f, 0.f);
}

// support = (1-ALPHA)*hi + ALPHA*h0, written in place over hi
__global__ __launch_bounds__(256)
void gcnii_combine_f4(float4* __restrict__ s, const float4* __restrict__ h0, int n4)
{
    const int i = blockIdx.x * 256 + threadIdx.x;
    if (i >= n4) return;
    float4 a = s[i];
    float4 b = h0[i];
    a.x = 0.9f * a.x + 0.1f * b.x;
    a.y = 0.9f * a.y + 0.1f * b.y;
    a.z = 0.9f * a.z + 0.1f * b.z;
    a.w = 0.9f * a.w + 0.1f * b.w;
    s[i] = a;
}

extern "C" void kernel_launch(void* const* d_in, const int* in_sizes, int n_in,
                              void* d_out, int out_size, void* d_ws, size_t ws_size,
                              hipStream_t stream)
{
    (void)n_in; (void)out_size; (void)ws_size;

    const float* features = (const float*)d_in[0];
    const int*   erow     = (const int*)d_in[1];
    const int*   ecol     = (const int*)d_in[2];
    const float* eval     = (const float*)d_in[3];
    const float* W_in     = (const float*)d_in[4];
    const float* b_in     = (const float*)d_in[5];
    const float* Wc       = (const float*)d_in[6];
    const float* W_out    = (const float*)d_in[7];
    const float* b_out    = (const float*)d_in[8];
    float*       out      = (float*)d_out;

    const int NFEAT = 512, NH = 256, NC = 40, NL = 8;
    const int M = in_sizes[0] / NFEAT;   // 100000
    const int E = in_sizes[1];           // 3200000

    // workspace: h | h0 | hi  (each M*NH floats = ~102.4 MB)
    float* h  = (float*)d_ws;
    float* h0 = h  + (size_t)M * NH;
    float* hi = h0 + (size_t)M * NH;

    dim3 blk(256);
    dim3 gridGemm((NH + 63) / 64, (M + 127) / 128);

    // h = relu(features @ W_in + b_in); dup-store into h0
    gcnii_wmma_gemm_f32<<<gridGemm, blk, 0, stream>>>(
        features, W_in, b_in, nullptr, h, h0, M, NFEAT, NH, 1.f, 0.f, 1);

    const int n4 = (M * NH) / 4;
    const int gridEw = (int)(((long long)E * 32 + 255) / 256);
    const int gridE4 = (n4 + 255) / 256;

    for (int i = 0; i < NL; ++i) {
        const float theta = logf(0.5f / (float)(i + 1) + 1.0f);
        gcnii_zero_f4<<<gridE4, blk, 0, stream>>>((float4*)hi, n4);
        gcnii_spmm_edges<<<gridEw, blk, 0, stream>>>(erow, ecol, eval, h, hi, E);
        gcnii_combine_f4<<<gridE4, blk, 0, stream>>>((float4*)hi, (const float4*)h0, n4);
        // h = relu(theta * (S @ Wc_i) + (1-theta) * S), S == hi
        gcnii_wmma_gemm_f32<<<gridGemm, blk, 0, stream>>>(
            hi, Wc + (size_t)i * NH * NH, nullptr, hi, h, nullptr,
            M, NH, NH, theta, 1.f - theta, 1);
    }

    // out = h @ W_out + b_out   (Ncols = 40, tail-guarded)
    dim3 gridOut(1, (M + 127) / 128);
    gcnii_wmma_gemm_f32<<<gridOut, blk, 0, stream>>>(
        h, W_out, b_out, nullptr, out, nullptr, M, NH, NC, 1.f, 0.f, 0);
}